// SemanticGraphConstructor_87634512708196
// MI455X (gfx1250) — compile-verified
//
#include <hip/hip_runtime.h>
#include <hip/hip_bf16.h>
#include <math.h>

// ---------------------------------------------------------------------------
// Problem constants (match reference)
// ---------------------------------------------------------------------------
#define C_CH   22
#define F_DIM  448
#define H_DIM  64
#define B_SZ   1024
#define P_PAIR 231                 // 22*21/2
#define ROWS   (B_SZ * C_CH)       // 22528
#define LN_EPS 1e-5f

typedef float v2f __attribute__((ext_vector_type(2)));
typedef float v8f __attribute__((ext_vector_type(8)));

// ---------------------------------------------------------------------------
// Kernel 0: init adjacency to identity (off-diagonals overwritten later)
// ---------------------------------------------------------------------------
__global__ void adj_init_kernel(float* __restrict__ out, int n) {
    int idx = blockIdx.x * blockDim.x + threadIdx.x;
    if (idx < n) {
        int r = idx % (C_CH * C_CH);
        out[idx] = ((r % (C_CH + 1)) == 0) ? 1.0f : 0.0f;
    }
}

// ---------------------------------------------------------------------------
// Kernel 1: E1[c][h] = emb[c] @ W1[0:64],  E2[c][h] = emb[c] @ W1[64:128]
// ---------------------------------------------------------------------------
__global__ void emb_proj_kernel(const float* __restrict__ emb,
                                const float* __restrict__ W1,
                                float* __restrict__ E1,
                                float* __restrict__ E2) {
    int c = blockIdx.x;
    int k = threadIdx.x;
    float s1 = 0.0f, s2 = 0.0f;
#pragma unroll 8
    for (int e = 0; e < H_DIM; ++e) {
        float ev = emb[c * H_DIM + e];
        s1 += ev * W1[e * H_DIM + k];
        s2 += ev * W1[(H_DIM + e) * H_DIM + k];
    }
    E1[c * H_DIM + k] = s1;
    E2[c * H_DIM + k] = s2;
}

// ---------------------------------------------------------------------------
// Kernel 2: UV[row][0:64]   = x[row] @ W1[128:576]   (U part)
//           UV[row][64:128] = x[row] @ W1[576:1024]  (V part)
// One wave per 16-row M-tile, all 8 N-tiles. f32 WMMA 16x16x4.
// B panel staged via LDS in 16-K chunks, K-pair interleaved so each
// B fragment is a single aligned ds_load_b64. Next chunk is prefetched
// into registers while the current chunk is consumed (software pipeline).
// ---------------------------------------------------------------------------
#define KCHUNK   16
#define KPAIRS   (KCHUNK / 2)          // 8 k-pair rows per chunk
#define KPSTRIDE 288                   // dwords per k-pair row: 128*2 + 32 pad

__global__ __launch_bounds__(256) void uv_gemm_kernel(
        const float* __restrict__ x,
        const float* __restrict__ W1,
        float* __restrict__ UV) {
    __shared__ float bsh[KPAIRS * KPSTRIDE];   // 9216 B

    const int lane = threadIdx.x & 31;
    const int wv   = threadIdx.x >> 5;
    const int mtile = blockIdx.x * 8 + wv;
    const int rowbase = mtile * 16;
    const int lm = lane & 15;      // M (or N) index within tile
    const int lh = lane >> 4;      // half-wave selects K pair

    v8f acc[8];
#pragma unroll
    for (int nt = 0; nt < 8; ++nt) acc[nt] = (v8f)(0.0f);

    // per-lane A stream: row rowbase+lm, starting K = lh*2
    const float* arow = x + (size_t)(rowbase + lm) * F_DIM + lh * 2;

    // ---- stage chunk 0 of B' into registers ----
    // B'[k][n] = n<64 ? W1[128+k][n] : W1[576+k][n-64]
    float breg[8];
#pragma unroll
    for (int r = 0; r < 8; ++r) {
        int idx = threadIdx.x + r * 256;
        int kk  = idx >> 7;            // 0..15
        int n   = idx & 127;           // coalesced along n
        int e   = ((n < 64) ? 128 : 576) + kk;
        breg[r] = W1[e * H_DIM + (n & 63)];
    }

    for (int k0 = 0; k0 < F_DIM; k0 += KCHUNK) {
        // ---- commit staged registers to LDS (k-pair interleaved) ----
#pragma unroll
        for (int r = 0; r < 8; ++r) {
            int idx = threadIdx.x + r * 256;
            int kk  = idx >> 7;
            int n   = idx & 127;
            bsh[(kk >> 1) * KPSTRIDE + n * 2 + (kk & 1)] = breg[r];
        }
        __syncthreads();

        // ---- stage next chunk into registers (overlaps with WMMA below) ----
        if (k0 + KCHUNK < F_DIM) {
#pragma unroll
            for (int r = 0; r < 8; ++r) {
                int idx = threadIdx.x + r * 256;
                int kk  = idx >> 7;
                int n   = idx & 127;
                int e   = ((n < 64) ? 128 : 576) + k0 + KCHUNK + kk;
                breg[r] = W1[e * H_DIM + (n & 63)];
            }
            // cache prefetch for the A stream of the next chunk
            __builtin_prefetch(arow + k0 + KCHUNK, 0, 1);
        }

        // ---- 4 K-steps x 8 N-tiles of f32 WMMA ----
#pragma unroll
        for (int kk = 0; kk < KCHUNK; kk += 4) {
            // A fragment 16x4 f32: lanes 0-15 -> K0/K1, lanes 16-31 -> K2/K3
            v2f a;
            a.x = arow[k0 + kk];
            a.y = arow[k0 + kk + 1];
            const float* bp = &bsh[((kk >> 1) + lh) * KPSTRIDE];
#pragma unroll
            for (int nt = 0; nt < 8; ++nt) {
                // single aligned ds_load_b64 per fragment
                v2f bb = *(const v2f*)(bp + (nt * 16 + lm) * 2);
                acc[nt] = __builtin_amdgcn_wmma_f32_16x16x4_f32(
                    /*neg_a=*/false, a, /*neg_b=*/false, bb,
                    /*c_mod=*/(short)0, acc[nt],
                    /*reuse_a=*/false, /*reuse_b=*/false);
            }
        }
        __syncthreads();
    }

    // C/D layout: VGPR g holds M=g (lanes 0-15) and M=g+8 (lanes 16-31)
#pragma unroll
    for (int nt = 0; nt < 8; ++nt) {
#pragma unroll
        for (int g = 0; g < 8; ++g) {
            int r = rowbase + lh * 8 + g;
            UV[(size_t)r * 128 + nt * 16 + lm] = acc[nt][g];
        }
    }
}

// ---------------------------------------------------------------------------
// Kernel 3: per-(batch, pair) edge MLP tail. One wave per edge.
//   h = U[b,i] + V[b,j] + E1[i] + E2[j] + b1  -> LN -> ReLU
//   h2 = ReLU(h @ W2 + b2); w = sigmoid(h2 @ W3 + b3)
//   adj[b,i,j] = adj[b,j,i] = (w > sigmoid(thr)) ? w : 0
// ---------------------------------------------------------------------------
__device__ __forceinline__ float wave_reduce_add(float v) {
#pragma unroll
    for (int off = 16; off >= 1; off >>= 1)
        v += __shfl_xor(v, off, 32);
    return v;
}

__global__ __launch_bounds__(256) void edge_mlp_kernel(
        const float* __restrict__ UV,
        const float* __restrict__ E1,
        const float* __restrict__ E2,
        const float* __restrict__ b1,
        const float* __restrict__ ln_g,
        const float* __restrict__ ln_b,
        const float* __restrict__ W2,
        const float* __restrict__ b2,
        const float* __restrict__ W3,
        const float* __restrict__ b3,
        const float* __restrict__ sparsity,
        float* __restrict__ out) {
    __shared__ float hsh[8][H_DIM];

    const int lane = threadIdx.x & 31;
    const int wv   = threadIdx.x >> 5;
    const int task = blockIdx.x * 8 + wv;        // 236544 tasks exactly
    const int b = task / P_PAIR;
    const int p = task % P_PAIR;

    // p -> (i, j) upper-triangle row-major (np.triu_indices order)
    int i = 0, rem = p, rl = C_CH - 1;
    while (rem >= rl) { rem -= rl; --rl; ++i; }
    int j = i + 1 + rem;

    const float* uvI = UV + (size_t)(b * C_CH + i) * 128;        // U part
    const float* uvJ = UV + (size_t)(b * C_CH + j) * 128 + 64;   // V part

    int k0 = lane, k1 = lane + 32;
    float v0 = uvI[k0] + uvJ[k0] + E1[i * H_DIM + k0] + E2[j * H_DIM + k0] + b1[k0];
    float v1 = uvI[k1] + uvJ[k1] + E1[i * H_DIM + k1] + E2[j * H_DIM + k1] + b1[k1];

    // LayerNorm over 64
    float mu = wave_reduce_add(v0 + v1) * (1.0f / 64.0f);
    float d0 = v0 - mu, d1 = v1 - mu;
    float var = wave_reduce_add(d0 * d0 + d1 * d1) * (1.0f / 64.0f);
    float rstd = rsqrtf(var + LN_EPS);
    float h0 = fmaxf(d0 * rstd * ln_g[k0] + ln_b[k0], 0.0f);
    float h1 = fmaxf(d1 * rstd * ln_g[k1] + ln_b[k1], 0.0f);

    hsh[wv][k0] = h0;
    hsh[wv][k1] = h1;
    __syncthreads();

    // h @ W2 (64x32): lane computes output column `lane`
    float o = b2[lane];
#pragma unroll 8
    for (int k = 0; k < H_DIM; ++k)
        o += hsh[wv][k] * W2[k * 32 + lane];
    float h2 = fmaxf(o, 0.0f);

    // h2 @ W3 (32x1) + sigmoid
    float z = wave_reduce_add(h2 * W3[lane]) + b3[0];
    float w = 1.0f / (1.0f + expf(-z));
    float thr = 1.0f / (1.0f + expf(-sparsity[0]));

    if (lane == 0) {
        float outv = (w > thr) ? w : 0.0f;
        out[(size_t)b * (C_CH * C_CH) + i * C_CH + j] = outv;
        out[(size_t)b * (C_CH * C_CH) + j * C_CH + i] = outv;
    }
}

// ---------------------------------------------------------------------------
// Host launcher
// ---------------------------------------------------------------------------
extern "C" void kernel_launch(void* const* d_in, const int* in_sizes, int n_in,
                              void* d_out, int out_size, void* d_ws, size_t ws_size,
                              hipStream_t stream) {
    const float* x    = (const float*)d_in[0];
    const float* emb  = (const float*)d_in[1];
    const float* W1   = (const float*)d_in[2];
    const float* b1   = (const float*)d_in[3];
    const float* ln_g = (const float*)d_in[4];
    const float* ln_b = (const float*)d_in[5];
    const float* W2   = (const float*)d_in[6];
    const float* b2   = (const float*)d_in[7];
    const float* W3   = (const float*)d_in[8];
    const float* b3   = (const float*)d_in[9];
    const float* thr  = (const float*)d_in[10];
    float* out = (float*)d_out;

    // Workspace layout: UV (22528x128 f32) | E1 (22x64) | E2 (22x64)
    float* UV = (float*)d_ws;
    float* E1 = UV + (size_t)ROWS * 128;
    float* E2 = E1 + C_CH * H_DIM;

    // 0) adjacency = I
    int n_adj = B_SZ * C_CH * C_CH;
    adj_init_kernel<<<(n_adj + 255) / 256, 256, 0, stream>>>(out, n_adj);

    // 1) embedding projections E1/E2
    emb_proj_kernel<<<C_CH, H_DIM, 0, stream>>>(emb, W1, E1, E2);

    // 2) main WMMA GEMM: 22528 x 448 x 128 -> UV
    uv_gemm_kernel<<<ROWS / 16 / 8, 256, 0, stream>>>(x, W1, UV);

    // 3) edge MLP tail: one wave per (batch, pair); 236544 / 8 = 29568 blocks
    edge_mlp_kernel<<<(B_SZ * P_PAIR) / 8, 256, 0, stream>>>(
        UV, E1, E2, b1, ln_g, ln_b, W2, b2, W3, b3, thr, out);
}